// HungarianMatcherGroup_13795434955234
// MI455X (gfx1250) — compile-verified
//
#include <hip/hip_runtime.h>

// ---------------------------------------------------------------------------
// HungarianMatcherGroup cost matrix on gfx1250 (CDNA5, wave32).
//
// cost_class is computed as an exact one-hot GEMM with V_WMMA_F32_16X16X4_F32:
//   F[N,80] (focal table) x OneHot[80,T]  ==  gather F[n, label[t]]
// bbox-L1 and GIoU terms are fused per D-matrix element, group-of-3 row max
// applied in LDS before a single coalesced store.
// ---------------------------------------------------------------------------

typedef __attribute__((ext_vector_type(2))) float v2f;
typedef __attribute__((ext_vector_type(8))) float v8f;

#define BSZ      16
#define QN       900
#define CCLS     80
#define TT       1600
#define GRP      3
#define NROWS    (BSZ * QN)      // 14400 pred rows
#define RB       48              // pred rows per block (divisible by GRP and 16)
#define ORB      (RB / GRP)      // 16 output rows per block
#define CB       160             // target cols per block (divisible by 16)
#define NTHREADS 256
#define NWAVES   (NTHREADS / 32)

#define ALPHA_F  0.25f
#define EPS_F    1e-8f
#define W_CLASS  2.0f
#define W_BBOX   5.0f
#define W_GIOU   2.0f

__global__ __launch_bounds__(NTHREADS)
void matcher_cost_kernel(const float* __restrict__ logits,   // [N, 80]
                         const float* __restrict__ pboxes,   // [N, 4] cxcywh
                         const int*   __restrict__ tlabels,  // [T]
                         const float* __restrict__ tboxes,   // [T, 4] cxcywh
                         float*       __restrict__ out)      // [N/GRP, T]
{
    __shared__ float Fl[RB][CCLS];     // focal table (pos - neg) per class
    __shared__ float predD[RB][10];    // cx,cy,w,h,x0,y0,x1,y1,area
    __shared__ float tgtD[CB][10];
    __shared__ int   labl[CB];
    __shared__ float Cfull[RB][CB];    // full cost tile before group-max

    const int tid     = threadIdx.x;
    const int rowbase = blockIdx.x * RB;   // 300 row blocks
    const int colbase = blockIdx.y * CB;   // 10 col blocks

    // ---- stage pred rows: cxcywh -> xyxy + area ----
    for (int r = tid; r < RB; r += NTHREADS) {
        const float4 b = *(const float4*)&pboxes[(size_t)(rowbase + r) * 4];
        const float x0 = b.x - 0.5f * b.z, y0 = b.y - 0.5f * b.w;
        const float x1 = b.x + 0.5f * b.z, y1 = b.y + 0.5f * b.w;
        predD[r][0] = b.x; predD[r][1] = b.y; predD[r][2] = b.z; predD[r][3] = b.w;
        predD[r][4] = x0;  predD[r][5] = y0;  predD[r][6] = x1;  predD[r][7] = y1;
        predD[r][8] = (x1 - x0) * (y1 - y0);
    }
    // ---- stage target cols ----
    for (int c = tid; c < CB; c += NTHREADS) {
        const int g = colbase + c;
        labl[c] = tlabels[g];
        const float4 b = *(const float4*)&tboxes[(size_t)g * 4];
        const float x0 = b.x - 0.5f * b.z, y0 = b.y - 0.5f * b.w;
        const float x1 = b.x + 0.5f * b.z, y1 = b.y + 0.5f * b.w;
        tgtD[c][0] = b.x; tgtD[c][1] = b.y; tgtD[c][2] = b.z; tgtD[c][3] = b.w;
        tgtD[c][4] = x0;  tgtD[c][5] = y0;  tgtD[c][6] = x1;  tgtD[c][7] = y1;
        tgtD[c][8] = (x1 - x0) * (y1 - y0);
    }
    // ---- focal-cost table: Fl[r][c] = pos - neg ----
    for (int i = tid; i < RB * CCLS; i += NTHREADS) {
        const int r = i / CCLS, c = i % CCLS;
        const float x   = logits[(size_t)(rowbase + r) * CCLS + c];
        const float p   = 1.0f / (1.0f + __expf(-x));
        const float neg = (1.0f - ALPHA_F) * p * p * (-__logf(1.0f - p + EPS_F));
        const float pos = ALPHA_F * (1.0f - p) * (1.0f - p) * (-__logf(p + EPS_F));
        Fl[r][c] = pos - neg;
    }
    __syncthreads();

    // ---- WMMA tiles: 3 x 10 = 30 tiles of 16x16, wave-uniform loop ----
    const int lane = tid & 31;
    const int wave = tid >> 5;
    const int ml   = lane & 15;   // M for A-matrix / N for B,C,D
    const int hi   = lane >> 4;

    const int NT_C = CB / 16;
    for (int tile = wave; tile < (RB / 16) * NT_C; tile += NWAVES) {
        const int r0 = (tile / NT_C) * 16;
        const int c0 = (tile % NT_C) * 16;

        // one-hot GEMM over K = 80 classes (20 chunks of K=4)
        v8f acc = {};
        const int am  = r0 + ml;          // A row (pred)
        const int lab = labl[c0 + ml];    // B column's label
        for (int k0 = 0; k0 < CCLS; k0 += 4) {
            const int k = k0 + 2 * hi;    // this lane's K pair
            v2f a; a.x = Fl[am][k];       a.y = Fl[am][k + 1];
            v2f b; b.x = (lab == k    ) ? 1.0f : 0.0f;
                   b.y = (lab == k + 1) ? 1.0f : 0.0f;
            acc = __builtin_amdgcn_wmma_f32_16x16x4_f32(
                      false, a, false, b, (short)0, acc, false, false);
        }

        // fuse bbox-L1 + GIoU per D element: elem j -> M = j + 8*hi, N = ml
        const int cc = c0 + ml;
        const float tcx = tgtD[cc][0], tcy = tgtD[cc][1];
        const float twd = tgtD[cc][2], tht = tgtD[cc][3];
        const float tx0 = tgtD[cc][4], ty0 = tgtD[cc][5];
        const float tx1 = tgtD[cc][6], ty1 = tgtD[cc][7];
        const float tar = tgtD[cc][8];
#pragma unroll
        for (int j = 0; j < 8; ++j) {
            const int rr = r0 + j + 8 * hi;
            const float pcx = predD[rr][0], pcy = predD[rr][1];
            const float pwd = predD[rr][2], pht = predD[rr][3];
            const float px0 = predD[rr][4], py0 = predD[rr][5];
            const float px1 = predD[rr][6], py1 = predD[rr][7];
            const float par = predD[rr][8];

            const float bbox = fabsf(pcx - tcx) + fabsf(pcy - tcy) +
                               fabsf(pwd - twd) + fabsf(pht - tht);

            const float iw    = fmaxf(fminf(px1, tx1) - fmaxf(px0, tx0), 0.0f);
            const float ih    = fmaxf(fminf(py1, ty1) - fmaxf(py0, ty0), 0.0f);
            const float inter = iw * ih;
            const float uni   = par + tar - inter;
            const float iou   = inter / uni;

            const float ew    = fmaxf(fmaxf(px1, tx1) - fminf(px0, tx0), 0.0f);
            const float eh    = fmaxf(fmaxf(py1, ty1) - fminf(py0, ty0), 0.0f);
            const float ea    = ew * eh;
            const float giou  = iou - (ea - uni) / ea;

            Cfull[rr][cc] = W_BBOX * bbox + W_CLASS * acc[j] - W_GIOU * giou;
        }
    }
    __syncthreads();

    // ---- group-of-3 max + coalesced store ----
    const int outRowBase = rowbase / GRP;
    for (int i = tid; i < ORB * CB; i += NTHREADS) {
        const int r = i / CB, c = i % CB;
        const float v = fmaxf(fmaxf(Cfull[GRP * r][c], Cfull[GRP * r + 1][c]),
                              Cfull[GRP * r + 2][c]);
        out[(size_t)(outRowBase + r) * TT + colbase + c] = v;
    }
}

extern "C" void kernel_launch(void* const* d_in, const int* in_sizes, int n_in,
                              void* d_out, int out_size, void* d_ws, size_t ws_size,
                              hipStream_t stream) {
    const float* logits  = (const float*)d_in[0];  // [16,900,80]
    const float* pboxes  = (const float*)d_in[1];  // [16,900,4]
    const int*   tlabels = (const int*)  d_in[2];  // [1600]
    const float* tboxes  = (const float*)d_in[3];  // [1600,4]
    // d_in[4] = g_size (3), baked into GRP
    float* out = (float*)d_out;                    // [16,300,1600]

    dim3 grid(NROWS / RB, TT / CB);                // (300, 10)
    matcher_cost_kernel<<<grid, dim3(NTHREADS), 0, stream>>>(
        logits, pboxes, tlabels, tboxes, out);
}